// EEGGraphNet_23158463660630
// MI455X (gfx1250) — compile-verified
//
#include <hip/hip_runtime.h>
#include <math.h>

// ---------------------------------------------------------------------------
// EEG GraphSAGE forward, fp32, WMMA f32 16x16x4 for all node-level GEMMs.
// N=50000, E=800000, F_IN=128, H=64, G=64, C=4.
// ---------------------------------------------------------------------------

typedef float v2f __attribute__((ext_vector_type(2)));
typedef float v8f __attribute__((ext_vector_type(8)));

#define BN_EPS 1e-5f

__device__ __forceinline__ float elu1(float y) { return y > 0.f ? y : expm1f(y); }

// Fire-and-forget float add -> native GLOBAL_ATOMIC_ADD_F32 (no return value,
// STOREcnt path; avoids any CAS-loop fallback of generic atomicAdd).
__device__ __forceinline__ void atomicAddF(float* p, float v) {
  __hip_atomic_fetch_add(p, v, __ATOMIC_RELAXED, __HIP_MEMORY_SCOPE_AGENT);
}

// ------------------------------- init kernels ------------------------------

__global__ void fill_kernel(float* __restrict__ p, float v, int n) {
  int t = blockIdx.x * blockDim.x + threadIdx.x;
  if (t < n) p[t] = v;
}

// emb layout: [g][0..63] = mean accum (0), [g][64..127] = max accum (-inf)
__global__ void init_emb_kernel(float* __restrict__ emb, int n) {
  int t = blockIdx.x * blockDim.x + threadIdx.x;
  if (t < n) emb[t] = ((t & 127) < 64) ? 0.f : -INFINITY;
}

__global__ void deg_kernel(const int* __restrict__ dst, float* __restrict__ deg, int E) {
  int t = blockIdx.x * blockDim.x + threadIdx.x;
  if (t < E) atomicAddF(&deg[dst[t]], 1.f);
}

__global__ void gcnt_kernel(const int* __restrict__ batch, float* __restrict__ gcnt, int n) {
  int t = blockIdx.x * blockDim.x + threadIdx.x;
  if (t < n) atomicAddF(&gcnt[batch[t]], 1.f);
}

// ------------------------- edge scatter (mean-agg sum) ---------------------
// One thread = one edge x 4 consecutive feature columns: float4 gather from
// h[src], 4x GLOBAL_ATOMIC_ADD_F32 into agg[dst].

template <int DIM>
__global__ void scatter_add_kernel(const float* __restrict__ h, const int* __restrict__ src,
                                   const int* __restrict__ dst, float* __restrict__ agg, int E) {
  constexpr int D4 = DIM / 4;
  int t = blockIdx.x * blockDim.x + threadIdx.x;
  int e = t / D4;
  if (e >= E) return;
  int c = (t - e * D4) * 4;
  int s = src[e];
  int d = dst[e];
  const float4 v = *(const float4*)(h + (size_t)s * DIM + c);
  float* base = agg + (size_t)d * DIM + c;
  atomicAddF(base + 0, v.x);
  atomicAddF(base + 1, v.y);
  atomicAddF(base + 2, v.z);
  atomicAddF(base + 3, v.w);
}

// --------------------- fused SAGE layer: GEMM + BN + ELU -------------------
// hout = ELU(BN( (agg*inv_deg) @ Wl + hin @ Wr + bias ))
// Single GEMM over concatenated operand A = [agg*inv_deg | hin] (16 x 2*DIN
// tile staged in LDS, +4 padding => conflict-free column reads).  B = [Wl;Wr]
// read from global (<= 64 KB, L2/WGP$-resident).  One wave = one 16x16
// tile; 4 waves/block cover all 64 output columns; block handles 16 nodes.
// K-loop is split into a Wl pass and a Wr pass => branch-free inner loop:
// ds_load_b64 (A) + 2x global_load_b32 (B) + v_wmma_f32_16x16x4_f32.

template <int DIN>
__global__ __launch_bounds__(128) void sage_layer_kernel(
    const float* __restrict__ agg, const float* __restrict__ hin,
    const float* __restrict__ deg, const float* __restrict__ Wl,
    const float* __restrict__ Wr, const float* __restrict__ bias,
    const float* __restrict__ bng, const float* __restrict__ bnb,
    const float* __restrict__ bnm, const float* __restrict__ bnv,
    float* __restrict__ hout, int nnodes) {
  constexpr int K   = 2 * DIN;
  constexpr int LDK = K + 4;  // pad: row stride = 4 banks -> conflict-free
  __shared__ float As[16 * LDK];

  const int m0 = blockIdx.x * 16;

  // cooperative A-tile load: 16 rows x K cols, float4 granularity
  for (int i = threadIdx.x; i < 4 * K; i += 128) {  // 16*K/4 float4 chunks
    int row  = i / (K / 4);
    int c4   = (i - row * (K / 4)) * 4;
    int node = m0 + row;
    if (node >= nnodes) node = nnodes - 1;
    float4 v;
    if (c4 < DIN) {
      v = *(const float4*)(agg + (size_t)node * DIN + c4);
      float inv = 1.f / fmaxf(deg[node], 1.f);
      v.x *= inv; v.y *= inv; v.z *= inv; v.w *= inv;
    } else {
      v = *(const float4*)(hin + (size_t)node * DIN + (c4 - DIN));
    }
    *(float4*)(&As[row * LDK + c4]) = v;
  }
  __syncthreads();

  const int lane  = threadIdx.x & 31;
  const int wave  = threadIdx.x >> 5;
  const int mloc  = lane & 15;
  const int khalf = lane >> 4;          // 0: K={k,k+1}, 1: K={k+2,k+3}
  const int col   = wave * 16 + mloc;   // output column, 0..63

  // A-frag per ISA: lane m+16*(k/2) holds A[m][k], A[m][k+1]
  const float* As_lane = &As[mloc * LDK + 2 * khalf];  // 8B aligned (even off)
  v8f c = {};

  // pass 1: agg*inv_deg @ Wl
  {
    const float* b0 = Wl + 2 * khalf * 64 + col;
#pragma unroll 4
    for (int k = 0; k < DIN; k += 4) {
      v2f a = *(const v2f*)(As_lane + k);
      v2f b;
      b.x = b0[0];
      b.y = b0[64];
      b0 += 256;  // 4 weight rows per step
      c = __builtin_amdgcn_wmma_f32_16x16x4_f32(false, a, false, b, (short)0,
                                                c, false, false);
    }
  }
  // pass 2: hin @ Wr
  {
    const float* b0 = Wr + 2 * khalf * 64 + col;
    const float* a0 = As_lane + DIN;
#pragma unroll 4
    for (int k = 0; k < DIN; k += 4) {
      v2f a = *(const v2f*)(a0 + k);
      v2f b;
      b.x = b0[0];
      b.y = b0[64];
      b0 += 256;
      c = __builtin_amdgcn_wmma_f32_16x16x4_f32(false, a, false, b, (short)0,
                                                c, false, false);
    }
  }

  // epilogue: +bias, inference BN, ELU
  const float scale = bng[col] * rsqrtf(bnv[col] + BN_EPS);
  const float shift = bnb[col] + (bias[col] - bnm[col]) * scale;
#pragma unroll
  for (int r = 0; r < 8; r++) {
    int node = m0 + r + 8 * khalf;  // C layout: VGPR r = M=r (lanes 0-15), M=r+8 (16-31)
    if (node < nnodes) {
      float y = c[r] * scale + shift;
      hout[(size_t)node * 64 + col] = elu1(y);
    }
  }
}

// ------------------------------ graph pooling ------------------------------

__global__ void pool_kernel(const float* __restrict__ h, const int* __restrict__ batch,
                            float* __restrict__ emb, int n) {
  int t = blockIdx.x * blockDim.x + threadIdx.x;
  if (t >= n * 64) return;
  int node = t >> 6;
  int j    = t & 63;
  float val = h[t];
  int g = batch[node];
  atomicAddF(&emb[g * 128 + j], val);
  float* mp = &emb[g * 128 + 64 + j];
  if (val >= 0.f) atomicMax((int*)mp, __float_as_int(val));
  else            atomicMin((unsigned int*)mp, __float_as_uint(val));
}

// ------------------------------- readout head ------------------------------

__global__ void head1_kernel(const float* __restrict__ emb, const float* __restrict__ gcnt,
                             const float* __restrict__ Wh1, const float* __restrict__ bh1,
                             const float* __restrict__ bng, const float* __restrict__ bnb,
                             const float* __restrict__ bnm, const float* __restrict__ bnv,
                             float* __restrict__ z) {
  int t = blockIdx.x * blockDim.x + threadIdx.x;  // G*H = 4096
  if (t >= 64 * 64) return;
  int g = t >> 6;
  int j = t & 63;
  float inv = 1.f / fmaxf(gcnt[g], 1.f);
  float acc = bh1[j];
  for (int k = 0; k < 128; k++) {
    float e = emb[g * 128 + k];
    if (k < 64) e *= inv;  // finalize mean pool on the fly
    acc += e * Wh1[k * 64 + j];
  }
  float y = (acc - bnm[j]) * rsqrtf(bnv[j] + BN_EPS) * bng[j] + bnb[j];
  z[t] = elu1(y);
}

__global__ void head2_kernel(const float* __restrict__ z, const float* __restrict__ Wh2,
                             const float* __restrict__ bh2, float* __restrict__ out) {
  int t = blockIdx.x * blockDim.x + threadIdx.x;  // G*C = 256
  if (t >= 64 * 4) return;
  int g = t >> 2;
  int cc = t & 3;
  float acc = bh2[cc];
  for (int j = 0; j < 64; j++) acc += z[g * 64 + j] * Wh2[j * 4 + cc];
  out[t] = acc;
}

// ------------------------------- launcher ----------------------------------

extern "C" void kernel_launch(void* const* d_in, const int* in_sizes, int n_in,
                              void* d_out, int out_size, void* d_ws, size_t ws_size,
                              hipStream_t stream) {
  const float* x     = (const float*)d_in[0];
  const int*   ei    = (const int*)d_in[1];
  const int*   batch = (const int*)d_in[2];
  const int N = in_sizes[0] / 128;
  const int E = in_sizes[1] / 2;
  const int* src = ei;
  const int* dst = ei + E;

  // params in setup_inputs() insertion order
  const float* P[29];
  for (int i = 0; i < 29; i++) P[i] = (const float*)d_in[3 + i];
  // P[0..6]=layer0 {Wl,Wr,b,g,b,m,v}, P[7..13]=layer1, P[14..20]=layer2
  // P[21]=Wh1 P[22]=bh1 P[23..26]=bnh{g,b,m,v} P[27]=Wh2 P[28]=bh2

  // workspace carve-out (~52 MB)
  size_t off = 0;
  auto ws = [&](size_t nflt) -> float* {
    float* p = (float*)((char*)d_ws + off);
    off += ((nflt * sizeof(float) + 255) / 256) * 256;
    return p;
  };
  float* deg  = ws(N);
  float* gcnt = ws(64);
  float* emb  = ws(64 * 128);
  float* zb   = ws(64 * 64);
  float* agg  = ws((size_t)N * 128);
  float* hA   = ws((size_t)N * 64);
  float* hB   = ws((size_t)N * 64);
  (void)ws_size; (void)n_in; (void)out_size;

  const dim3 B256(256);
  auto nb = [](long long n) { return dim3((unsigned)((n + 255) / 256)); };
  const dim3 gemmGrid((N + 15) / 16), gemmBlk(128);

  // --- init + degree / graph counts ---
  fill_kernel<<<nb(N), B256, 0, stream>>>(deg, 0.f, N);
  fill_kernel<<<nb(64), B256, 0, stream>>>(gcnt, 0.f, 64);
  init_emb_kernel<<<nb(64 * 128), B256, 0, stream>>>(emb, 64 * 128);
  fill_kernel<<<nb((long long)N * 128), B256, 0, stream>>>(agg, 0.f, N * 128);
  deg_kernel<<<nb(E), B256, 0, stream>>>(dst, deg, E);
  gcnt_kernel<<<nb(N), B256, 0, stream>>>(batch, gcnt, N);

  // --- layer 0 (din=128) ---
  scatter_add_kernel<128><<<nb((long long)E * 32), B256, 0, stream>>>(x, src, dst, agg, E);
  sage_layer_kernel<128><<<gemmGrid, gemmBlk, 0, stream>>>(
      agg, x, deg, P[0], P[1], P[2], P[3], P[4], P[5], P[6], hA, N);

  // --- layer 1 (din=64) ---
  fill_kernel<<<nb((long long)N * 64), B256, 0, stream>>>(agg, 0.f, N * 64);
  scatter_add_kernel<64><<<nb((long long)E * 16), B256, 0, stream>>>(hA, src, dst, agg, E);
  sage_layer_kernel<64><<<gemmGrid, gemmBlk, 0, stream>>>(
      agg, hA, deg, P[7], P[8], P[9], P[10], P[11], P[12], P[13], hB, N);

  // --- layer 2 (din=64) ---
  fill_kernel<<<nb((long long)N * 64), B256, 0, stream>>>(agg, 0.f, N * 64);
  scatter_add_kernel<64><<<nb((long long)E * 16), B256, 0, stream>>>(hB, src, dst, agg, E);
  sage_layer_kernel<64><<<gemmGrid, gemmBlk, 0, stream>>>(
      agg, hB, deg, P[14], P[15], P[16], P[17], P[18], P[19], P[20], hA, N);

  // --- pooling + head ---
  pool_kernel<<<nb((long long)N * 64), B256, 0, stream>>>(hA, batch, emb, N);
  head1_kernel<<<nb(64 * 64), B256, 0, stream>>>(emb, gcnt, P[21], P[22], P[23], P[24],
                                                 P[25], P[26], zb);
  head2_kernel<<<nb(64 * 4), B256, 0, stream>>>(zb, P[27], P[28], (float*)d_out);
}